// Density_Softmax_20684562497972
// MI455X (gfx1250) — compile-verified
//
#include <hip/hip_runtime.h>
#include <hip/hip_bf16.h>
#include <float.h>
#include <math.h>

// Problem constants
constexpr int kB = 32;
constexpr int kC = 8192;
constexpr int kD = 256;
constexpr int kTOPK = 512;
constexpr int kSC = 32;            // C-subchunks for the big scan
constexpr int kCPS = kC / kSC;     // 256 classes per subchunk

typedef float v2f __attribute__((ext_vector_type(2)));
typedef float v8f __attribute__((ext_vector_type(8)));

// Workspace layout (in floats)
constexpr size_t OFF_WNORM = 0;                       // kC
constexpr size_t OFF_DIS   = OFF_WNORM + kC;          // kB*kC
constexpr size_t OFF_TDET  = OFF_DIS + (size_t)kB*kC; // kB*kD
constexpr size_t OFF_TERM  = OFF_TDET + (size_t)kB*kD;// kB*kD
constexpr size_t OFF_PART  = OFF_TERM + (size_t)kB*kD;
constexpr int    kNP       = kB * kSC * kD;           // partial slots (262144)
// partial array slots (each kNP elements, float or int)
constexpr int P_S = 0, P_M1 = 1, P_M2 = 2, P_NT1 = 3, P_NT2 = 4,
              P_MX1 = 5, P_MX2 = 6, P_MNF = 7, P_MXF = 8,
              P_I1 = 9, P_INT1 = 10, P_IMX1 = 11, P_CNT = 12;

// ---------------- kernel 1: per-class squared norms -----------------
__global__ __launch_bounds__(256) void k_wnorm(const float* __restrict__ w,
                                               float* __restrict__ wnorm) {
  int wave = threadIdx.x >> 5, lane = threadIdx.x & 31;
  int c = blockIdx.x * 8 + wave;
  const float* row = w + (size_t)c * kD;
  float s = 0.f;
  for (int i = lane; i < kD; i += 32) { float v = row[i]; s += v * v; }
  for (int off = 16; off; off >>= 1) s += __shfl_xor(s, off, 32);
  if (lane == 0) wnorm[c] = s;
}

// ---------------- kernel 2: WMMA distance GEMM ----------------------
// dis[b,c] = |s_b|^2 - 2 * <s_b, w_c> + |w_c|^2, s_b = w[labels[b]]
// One wave computes one 16x16 tile via V_WMMA_F32_16X16X4_F32, K-loop of 64.
__global__ __launch_bounds__(256) void k_dis(const float* __restrict__ w,
                                             const int* __restrict__ labels,
                                             const float* __restrict__ wnorm,
                                             float* __restrict__ dis) {
  int wave = threadIdx.x >> 5, lane = threadIdx.x & 31;
  int tile = blockIdx.x * 8 + wave;       // 0 .. (2*512 - 1)
  int tm = tile & 1;                      // B tile (2 of them)
  int tn = tile >> 1;                     // C tile (512 of them)
  int half = lane >> 4;                   // K sub-block select
  int l15 = lane & 15;

  int brow = tm * 16 + l15;               // A-matrix row (batch index)
  int ccol = tn * 16 + l15;               // B-matrix col (class index)
  const float* arow = w + (size_t)labels[brow] * kD;
  const float* wrow = w + (size_t)ccol * kD;

  v8f acc = {0.f, 0.f, 0.f, 0.f, 0.f, 0.f, 0.f, 0.f};
  for (int kk = 0; kk < kD; kk += 4) {
    int k = kk + half * 2;
    v2f a = *reinterpret_cast<const v2f*>(arow + k);  // A: 16x4 f32 layout
    v2f bv = *reinterpret_cast<const v2f*>(wrow + k); // B: 4x16 f32 layout
    acc = __builtin_amdgcn_wmma_f32_16x16x4_f32(
        false, a, false, bv, (short)0, acc, false, false);
  }

  // D tile: VGPR r holds M = r + half*8, N = l15
  float wn = wnorm[ccol];
#pragma unroll
  for (int r = 0; r < 8; ++r) {
    int bg = tm * 16 + half * 8 + r;
    float sn = wnorm[labels[bg]];
    dis[(size_t)bg * kC + ccol] = sn - 2.0f * acc[r] + wn;
  }
}

// ---------------- kernel 3: top-512 select + total_det --------------
__global__ __launch_bounds__(256) void k_select(const float* __restrict__ w,
                                                const float* __restrict__ mu,
                                                const float* __restrict__ var,
                                                const float* __restrict__ dis,
                                                float* __restrict__ tdet) {
  __shared__ unsigned keys[kC];   // 32 KB
  __shared__ int cnt;
  int b = blockIdx.x, tid = threadIdx.x;

  for (int i = tid; i < kC; i += 256) {
    unsigned bits = __float_as_uint(dis[(size_t)b * kC + i]);
    // order-preserving unsigned transform
    keys[i] = (bits & 0x80000000u) ? ~bits : (bits | 0x80000000u);
  }
  __syncthreads();

  // binary search for the 512th smallest key
  unsigned lo = 0u, hi = 0xFFFFFFFFu;
  while (lo < hi) {
    unsigned mid = lo + ((hi - lo) >> 1);
    if (tid == 0) cnt = 0;
    __syncthreads();
    int local = 0;
    for (int i = tid; i < kC; i += 256) local += (keys[i] <= mid) ? 1 : 0;
    atomicAdd(&cnt, local);
    __syncthreads();
    int total = cnt;
    __syncthreads();
    if (total >= kTOPK) hi = mid; else lo = mid + 1;
  }
  unsigned T = hi;

  // accumulate detached density over selected classes; thread = d
  float m = mu[b * kD + tid];
  float inv2v = 1.0f / (2.0f * var[b * kD + tid]);
  float acc = 0.f;
  for (int c = 0; c < kC; ++c) {
    if (keys[c] <= T) {
      float df = w[(size_t)c * kD + tid] - m;
      acc += __expf(-df * df * inv2v);
    }
  }
  tdet[b * kD + tid] = fmaxf(acc, 1e-8f);
}

// ---------------- kernel 4: streaming scan over all_class_density ---
// Block = 4 groups of 64 lanes; each group scans one 256-class subchunk.
// Each thread owns 4 contiguous d columns: one b128 + one b32 load per class.
__global__ __launch_bounds__(256) void k_scan(const float* __restrict__ acd,
                                              const unsigned char* __restrict__ ntb,
                                              float* __restrict__ ws) {
  int b  = blockIdx.x;
  int sc = blockIdx.y * 4 + (threadIdx.x >> 6);   // subchunk 0..31
  int d0 = (threadIdx.x & 63) * 4;                // base d (16B aligned)
  int c0 = sc * kCPS;
  size_t idx = ((size_t)b * kC + c0) * kD + d0;

  float s[4], m1[4], m2[4], nt1[4], nt2[4], mx1[4], mx2[4], mnF[4], mxF[4];
  int i1[4], in1[4], im1[4], cnt[4];
#pragma unroll
  for (int j = 0; j < 4; ++j) {
    s[j] = 0.f;
    m1[j] = FLT_MAX;  m2[j] = FLT_MAX;   i1[j] = -1;
    nt1[j] = FLT_MAX; nt2[j] = FLT_MAX;  in1[j] = -1;
    mx1[j] = -FLT_MAX; mx2[j] = -FLT_MAX; im1[j] = -1;
    mnF[j] = FLT_MAX; mxF[j] = -FLT_MAX; cnt[j] = 0;
  }

  for (int cc = 0; cc < kCPS; ++cc) {
    float4 v4 = *reinterpret_cast<const float4*>(acd + idx);
    unsigned ntw = *reinterpret_cast<const unsigned*>(ntb + idx);
    int c = c0 + cc;
    float vv[4] = {v4.x, v4.y, v4.z, v4.w};
#pragma unroll
    for (int j = 0; j < 4; ++j) {
      float v = vv[j];
      int nt = (ntw >> (8 * j)) & 0xFF;
      s[j] += v;
      if (v < m1[j]) { m2[j] = m1[j]; m1[j] = v; i1[j] = c; }
      else           { m2[j] = fminf(m2[j], v); }
      if (nt) {
        ++cnt[j];
        if (v < nt1[j]) { nt2[j] = nt1[j]; nt1[j] = v; in1[j] = c; }
        else            { nt2[j] = fminf(nt2[j], v); }
        if (v > mx1[j]) { mx2[j] = mx1[j]; mx1[j] = v; im1[j] = c; }
        else            { mx2[j] = fmaxf(mx2[j], v); }
      } else {
        mnF[j] = fminf(mnF[j], v);
        mxF[j] = fmaxf(mxF[j], v);
      }
    }
    idx += kD;
  }

  float* pf = ws + OFF_PART;
  int*   pi = (int*)(ws + OFF_PART);
#pragma unroll
  for (int j = 0; j < 4; ++j) {
    int p = (b * kSC + sc) * kD + (d0 + j);
    pf[P_S  * kNP + p] = s[j];
    pf[P_M1 * kNP + p] = m1[j];   pf[P_M2 * kNP + p] = m2[j];
    pf[P_NT1* kNP + p] = nt1[j];  pf[P_NT2* kNP + p] = nt2[j];
    pf[P_MX1* kNP + p] = mx1[j];  pf[P_MX2* kNP + p] = mx2[j];
    pf[P_MNF* kNP + p] = mnF[j];  pf[P_MXF* kNP + p] = mxF[j];
    pi[P_I1 * kNP + p] = i1[j];   pi[P_INT1*kNP + p] = in1[j];
    pi[P_IMX1*kNP + p] = im1[j];  pi[P_CNT* kNP + p] = cnt[j];
  }
}

// ---------------- kernel 5: merge partials -> per-(b,d) term --------
__global__ __launch_bounds__(256) void k_merge(const float* __restrict__ w,
                                               const float* __restrict__ mu,
                                               const float* __restrict__ var,
                                               const unsigned char* __restrict__ ntb,
                                               const int* __restrict__ labels,
                                               float* __restrict__ ws) {
  int t = blockIdx.x * 256 + threadIdx.x;  // 0 .. kB*kD-1
  int b = t >> 8, d = t & 255;
  const float* pf = ws + OFF_PART;
  const int*   pi = (const int*)(ws + OFF_PART);
  int base = (b * kSC) * kD + d;

  float s   = pf[P_S  * kNP + base];
  float m1  = pf[P_M1 * kNP + base], m2  = pf[P_M2 * kNP + base];
  float nt1 = pf[P_NT1* kNP + base], nt2 = pf[P_NT2* kNP + base];
  float mx1 = pf[P_MX1* kNP + base], mx2 = pf[P_MX2* kNP + base];
  float mnF = pf[P_MNF* kNP + base], mxF = pf[P_MXF* kNP + base];
  int i1 = pi[P_I1 * kNP + base], in1 = pi[P_INT1 * kNP + base];
  int im1 = pi[P_IMX1 * kNP + base], cntNT = pi[P_CNT * kNP + base];

  for (int ch = 1; ch < kSC; ++ch) {
    int p = base + ch * kD;
    s += pf[P_S * kNP + p];
    float bm1 = pf[P_M1 * kNP + p], bm2 = pf[P_M2 * kNP + p];
    if (bm1 < m1) { m2 = fminf(m1, bm2); m1 = bm1; i1 = pi[P_I1 * kNP + p]; }
    else          { m2 = fminf(m2, bm1); }
    float bn1 = pf[P_NT1 * kNP + p], bn2 = pf[P_NT2 * kNP + p];
    if (bn1 < nt1) { nt2 = fminf(nt1, bn2); nt1 = bn1; in1 = pi[P_INT1 * kNP + p]; }
    else           { nt2 = fminf(nt2, bn1); }
    float bx1 = pf[P_MX1 * kNP + p], bx2 = pf[P_MX2 * kNP + p];
    if (bx1 > mx1) { mx2 = fmaxf(mx1, bx2); mx1 = bx1; im1 = pi[P_IMX1 * kNP + p]; }
    else           { mx2 = fmaxf(mx2, bx1); }
    mnF = fminf(mnF, pf[P_MNF * kNP + p]);
    mxF = fmaxf(mxF, pf[P_MXF * kNP + p]);
    cntNT += pi[P_CNT * kNP + p];
  }

  float td = fmaxf(s, 1e-8f);
  int ntArg = (i1 >= 0) ? (int)ntb[((size_t)b * kC + i1) * kD + d] : 0;
  float nt_cnt = (float)(cntNT - (ntArg ? 1 : 0));
  // min over {nontrivial & c != argmin}; masked-out set gets +1000
  float minNTx = (in1 == i1) ? nt2 : nt1;
  float maxNTx = (im1 == i1) ? mx2 : mx1;
  float minv = fminf(minNTx, fminf(mnF, m1) + 1000.0f);
  float maxv = fmaxf(fmaxf(mxF, m1), maxNTx - 1000.0f);
  float cond = (minv - maxv >= 0.2f * td) ? 1.0f : 0.0f;
  float ov = cond * nt_cnt;  // = sum_c overly[b,c,d]

  float sw = w[(size_t)labels[b] * kD + d];
  float diff = sw - mu[t];
  float dens = __expf(-diff * diff / (2.0f * var[t]));
  float tdet = ws[OFF_TDET + t];
  ws[OFF_TERM + t] = dens / td * ((float)kC - ov) - dens / tdet * ov;
}

// ---------------- kernel 6: final reduction -------------------------
__global__ __launch_bounds__(256) void k_final(const float* __restrict__ ws,
                                               float* __restrict__ out) {
  __shared__ float red[8];
  float s = 0.f;
  for (int i = threadIdx.x; i < kB * kD; i += 256) s += ws[OFF_TERM + i];
  for (int off = 16; off; off >>= 1) s += __shfl_xor(s, off, 32);
  int wave = threadIdx.x >> 5, lane = threadIdx.x & 31;
  if (lane == 0) red[wave] = s;
  __syncthreads();
  if (threadIdx.x == 0) {
    float tot = 0.f;
    for (int i = 0; i < 8; ++i) tot += red[i];
    out[0] = tot / (float)((size_t)kB * kC * kD);
  }
}

extern "C" void kernel_launch(void* const* d_in, const int* in_sizes, int n_in,
                              void* d_out, int out_size, void* d_ws, size_t ws_size,
                              hipStream_t stream) {
  const float* w   = (const float*)d_in[0];        // weight [C,D]
  const float* mu  = (const float*)d_in[1];        // mu     [B,D]
  const float* var = (const float*)d_in[2];        // var    [B,D]
  const float* acd = (const float*)d_in[3];        // all_class_density [B,C,D]
  const int* labels = (const int*)d_in[4];         // labels [B]
  const unsigned char* ntb = (const unsigned char*)d_in[5]; // nontrivial [B,C,D]
  float* ws  = (float*)d_ws;
  float* out = (float*)d_out;

  k_wnorm <<<kC / 8, 256, 0, stream>>>(w, ws + OFF_WNORM);
  k_dis   <<<(kB / 16) * (kC / 16) / 8, 256, 0, stream>>>(w, labels,
                                                          ws + OFF_WNORM,
                                                          ws + OFF_DIS);
  k_select<<<kB, 256, 0, stream>>>(w, mu, var, ws + OFF_DIS, ws + OFF_TDET);
  k_scan  <<<dim3(kB, kSC / 4), 256, 0, stream>>>(acd, ntb, ws);
  k_merge <<<(kB * kD) / 256, 256, 0, stream>>>(w, mu, var, ntb, labels, ws);
  k_final <<<1, 256, 0, stream>>>(ws, out);
}